// Time_Series_Tokenization_33767032881250
// MI455X (gfx1250) — compile-verified
//
#include <hip/hip_runtime.h>
#include <math.h>

#define B_ 256
#define T_ 4096
#define C_ 16
#define PERIOD_ 12
#define HALF_ 6
#define NPB (T_ * C_)              /* 65536 elements per batch  */
#define XS_ROWS (T_ + 2 * HALF_)   /* 4108 rows incl. zero halo */

typedef float v2f __attribute__((ext_vector_type(2)));
typedef float v4f __attribute__((ext_vector_type(4)));
typedef float v8f __attribute__((ext_vector_type(8)));

// Band filter value: filt[d] for d in [0,12], [0.5, 1...1, 0.5]/12, else 0.
__device__ __forceinline__ float band_w(int d) {
    if (d == 0 || d == PERIOD_) return 0.5f / (float)PERIOD_;
    if (d > 0 && d < PERIOD_)   return 1.0f / (float)PERIOD_;
    return 0.0f;
}

__global__ __launch_bounds__(1024)
void stl_decompose_kernel(const float* __restrict__ dx,
                          const float* __restrict__ gmm,
                          const float* __restrict__ bta,
                          float* __restrict__ otr,
                          float* __restrict__ ose,
                          float* __restrict__ ore)
{
    extern __shared__ float xs[];            // XS_ROWS*16 floats: xs[(t+6)*16 + c]
    __shared__ float sums[PERIOD_ * C_];
    __shared__ float pavg[PERIOD_ * C_];
    __shared__ float mrow[C_];
    __shared__ float csum[C_];
    __shared__ float csq[C_];

    const int tid = threadIdx.x;
    const int b   = blockIdx.x;
    const size_t base = (size_t)b * NPB;

    v4f* xs4 = (v4f*)xs;

    // ---- init small LDS arrays + zero halo rows ----
    if (tid < PERIOD_ * C_) sums[tid] = 0.0f;
    if (tid < C_) { csum[tid] = 0.0f; csq[tid] = 0.0f; }
    const v4f vzero = {0.f, 0.f, 0.f, 0.f};
    if (tid < 24) {
        xs4[tid] = vzero;                                         // rows t=-6..-1
    } else if (tid < 48) {
        xs4[((T_ + HALF_) * C_) / 4 + (tid - 24)] = vzero;        // rows t=T..T+5
    }
    __syncthreads();

    // ---- load data_x -> LDS (b128, non-temporal), accumulate per-channel sum/sumsq ----
    const v4f* gx4 = (const v4f*)(dx + base);
    v4f ls = vzero;
    v4f lq = vzero;
    #pragma unroll
    for (int k = 0; k < 16; ++k) {
        int j = tid + 1024 * k;              // v4f index within the batch slab
        v4f v = __builtin_nontemporal_load(&gx4[j]);
        xs4[24 + j] = v;                     // 24 = 6 halo rows * 16 / 4
        ls += v;
        lq += v * v;
    }
    const int c0 = (tid << 2) & 15;          // this thread's 4 channels
    atomicAdd(&csum[c0 + 0], ls.x); atomicAdd(&csum[c0 + 1], ls.y);
    atomicAdd(&csum[c0 + 2], ls.z); atomicAdd(&csum[c0 + 3], ls.w);
    atomicAdd(&csq [c0 + 0], lq.x); atomicAdd(&csq [c0 + 1], lq.y);
    atomicAdd(&csq [c0 + 2], lq.z); atomicAdd(&csq [c0 + 3], lq.w);
    __syncthreads();

    // ---- m[c] = mean / sqrt(var), var with ddof=1 ----
    if (tid < C_) {
        float mean = csum[tid] * (1.0f / (float)T_);
        float var  = (csq[tid] - (float)T_ * mean * mean) * (1.0f / (float)(T_ - 1));
        mrow[tid]  = mean / sqrtf(var);
    }
    __syncthreads();

    // ---- transform in place: x = gamma*(raw - m) + beta ----
    {
        const v4f* gg4 = (const v4f*)(gmm + base);
        const v4f* bb4 = (const v4f*)(bta + base);
        v4f m4;
        m4.x = mrow[c0]; m4.y = mrow[c0 + 1]; m4.z = mrow[c0 + 2]; m4.w = mrow[c0 + 3];
        #pragma unroll
        for (int k = 0; k < 16; ++k) {
            int j = tid + 1024 * k;
            v4f v  = xs4[24 + j];
            v4f g  = __builtin_nontemporal_load(&gg4[j]);
            v4f be = __builtin_nontemporal_load(&bb4[j]);
            v = g * (v - m4) + be;
            xs4[24 + j] = v;
        }
    }
    __syncthreads();

    // ---- per-lane banded filter A (16x28 band, sliced into 7 K=4 WMMA chunks) ----
    // A layout (32-bit A 16x4): lanes 0-15 M=0..15; VGPR0: K=0 (lo half) / K=2 (hi half); VGPR1: K=1/K=3
    const int lane  = tid & 31;
    const int laneM = lane & 15;
    const int hi    = lane >> 4;
    v2f areg[7];
    #pragma unroll
    for (int s = 0; s < 7; ++s) {
        int k0 = 4 * s + 2 * hi;             // global K of slot .x in this slice
        areg[s].x = band_w(k0     - laneM);
        areg[s].y = band_w(k0 + 1 - laneM);
    }
    const int wave = tid >> 5;

    // Incremental phase tracking: t = t0 + r + 8*hi; t0 advances by 512 per tile
    // iteration and 512 % 12 == 8, so pbase = (t0 + 8*hi) % 12 is kept by +8 mod 12.
    int pbase0 = ((wave * 16) + 8 * hi) % PERIOD_;

    // ---- pass 1: trend via WMMA, write trend, accumulate per-phase detrended sums ----
    float* trb = otr + base;
    {
        int pbase = pbase0;
        for (int tile = wave; tile < T_ / 16; tile += 32) {
            const int t0 = tile * 16;
            v8f acc = {0.f, 0.f, 0.f, 0.f, 0.f, 0.f, 0.f, 0.f};
            #pragma unroll
            for (int s = 0; s < 7; ++s) {
                // B layout (4x16): VGPR0: lanes0-15 K=0 / lanes16-31 K=2; VGPR1: K=1/K=3; N=lane&15
                int rb = t0 + 4 * s + 2 * hi;    // xs row index == t0 + K (xs row r == x[t=r-6])
                v2f bv;
                bv.x = xs[rb * 16 + laneM];
                bv.y = xs[(rb + 1) * 16 + laneM];
                acc = __builtin_amdgcn_wmma_f32_16x16x4_f32(false, areg[s], false, bv,
                                                            (short)0, acc, false, false);
            }
            #pragma unroll
            for (int r = 0; r < 8; ++r) {
                int t = t0 + r + 8 * hi;         // D layout: VGPR r -> rows M=r (lo) / M=r+8 (hi)
                int p = pbase + r; p = (p >= PERIOD_) ? p - PERIOD_ : p;
                bool valid = (t >= HALF_) && (t < T_ - HALF_);
                float tv = valid ? acc[r] : 0.0f;
                __builtin_nontemporal_store(tv, &trb[t * 16 + laneM]);
                float d = valid ? (xs[(t + HALF_) * 16 + laneM] - acc[r]) : 0.0f;
                atomicAdd(&sums[p * C_ + laneM], d);   // masked to 0 outside valid range
            }
            pbase += 8; if (pbase >= PERIOD_) pbase -= PERIOD_;
        }
    }
    __syncthreads();

    // ---- pavg = sums/count, centered over phases ----
    if (tid < PERIOD_ * C_) {
        int p = tid >> 4;
        int first = HALF_ + ((p + HALF_) % PERIOD_);          // first valid t with phase p
        int cnt = (T_ - HALF_ - 1 - first) / PERIOD_ + 1;     // 341 for p in {6..9}, else 340
        pavg[tid] = sums[tid] / (float)cnt;
    }
    __syncthreads();
    if (tid < C_) {
        float mm = 0.0f;
        #pragma unroll
        for (int p = 0; p < PERIOD_; ++p) mm += pavg[p * C_ + tid];
        mm *= (1.0f / (float)PERIOD_);
        #pragma unroll
        for (int p = 0; p < PERIOD_; ++p) pavg[p * C_ + tid] -= mm;
    }
    __syncthreads();

    // ---- pass 2: recompute trend via WMMA, emit seasonal + resid ----
    float* seb = ose + base;
    float* reb = ore + base;
    {
        int pbase = pbase0;
        for (int tile = wave; tile < T_ / 16; tile += 32) {
            const int t0 = tile * 16;
            v8f acc = {0.f, 0.f, 0.f, 0.f, 0.f, 0.f, 0.f, 0.f};
            #pragma unroll
            for (int s = 0; s < 7; ++s) {
                int rb = t0 + 4 * s + 2 * hi;
                v2f bv;
                bv.x = xs[rb * 16 + laneM];
                bv.y = xs[(rb + 1) * 16 + laneM];
                acc = __builtin_amdgcn_wmma_f32_16x16x4_f32(false, areg[s], false, bv,
                                                            (short)0, acc, false, false);
            }
            #pragma unroll
            for (int r = 0; r < 8; ++r) {
                int t = t0 + r + 8 * hi;
                int p = pbase + r; p = (p >= PERIOD_) ? p - PERIOD_ : p;
                float xv = xs[(t + HALF_) * 16 + laneM];
                float sv = pavg[p * C_ + laneM];             // seasonal defined for ALL t
                bool valid = (t >= HALF_) && (t < T_ - HALF_);
                float rv = valid ? (xv - acc[r] - sv) : 0.0f;
                __builtin_nontemporal_store(sv, &seb[t * 16 + laneM]);
                __builtin_nontemporal_store(rv, &reb[t * 16 + laneM]);
            }
            pbase += 8; if (pbase >= PERIOD_) pbase -= PERIOD_;
        }
    }
}

extern "C" void kernel_launch(void* const* d_in, const int* in_sizes, int n_in,
                              void* d_out, int out_size, void* d_ws, size_t ws_size,
                              hipStream_t stream) {
    (void)in_sizes; (void)n_in; (void)out_size; (void)d_ws; (void)ws_size;
    const float* dx = (const float*)d_in[0];   // data_x
    // d_in[1] = data_y (unused by the reference)
    const float* gm = (const float*)d_in[2];   // gamma
    const float* bt = (const float*)d_in[3];   // beta
    float* out = (float*)d_out;
    float* otr = out;                                   // trend
    float* ose = out + (size_t)B_ * T_ * C_;            // seasonal
    float* ore = out + 2 * (size_t)B_ * T_ * C_;        // resid

    const size_t lds_bytes = (size_t)XS_ROWS * C_ * sizeof(float);  // 262,912 B (< 320 KB/WGP)
    stl_decompose_kernel<<<B_, 1024, lds_bytes, stream>>>(dx, gm, bt, otr, ose, ore);
}